// DCAB_81192061763903
// MI455X (gfx1250) — compile-verified
//
#include <hip/hip_runtime.h>
#include <cstddef>

// Problem constants
static constexpr int NB = 16;     // batch
static constexpr int NC = 256;    // channels
static constexpr int NH = 128;    // height
static constexpr int NW = 128;    // width
static constexpr int HWPIX = NH * NW;  // 16384
static constexpr int NK = 12;     // mixture kernels

typedef float v2f __attribute__((ext_vector_type(2)));
typedef float v4f __attribute__((ext_vector_type(4)));
typedef float v8f __attribute__((ext_vector_type(8)));

// ---------------------------------------------------------------------------
// Kernel 1: global average pool  y[b,c] = mean(x[b,c,:,:])
// One block per (b,c); 256 threads, 128-bit streaming NT loads (single pass
// over 268 MB with zero reuse -> don't pollute L2 ahead of the conv kernel).
// ---------------------------------------------------------------------------
__global__ __launch_bounds__(256) void k_gap(const float* __restrict__ x,
                                             float* __restrict__ y) {
  __shared__ float red[256];
  const int bc = blockIdx.x;
  const v4f* p4 = reinterpret_cast<const v4f*>(x + (size_t)bc * HWPIX);
  float s = 0.f;
  for (int i = threadIdx.x; i < HWPIX / 4; i += 256) {
    v4f v = __builtin_nontemporal_load(&p4[i]);
    s += v.x + v.y + v.z + v.w;
  }
  red[threadIdx.x] = s;
  __syncthreads();
  for (int off = 128; off > 0; off >>= 1) {
    if (threadIdx.x < off) red[threadIdx.x] += red[threadIdx.x + off];
    __syncthreads();
  }
  if (threadIdx.x == 0) y[bc] = red[0] * (1.0f / HWPIX);
}

// ---------------------------------------------------------------------------
// Kernel 2: ECA gate + gated pooled vector.
//   gate[b,c] = sigmoid(w0*y[b,c-1] + w1*y[b,c] + w2*y[b,c+1])  (zero pad in C)
//   p[b,c]    = gate[b,c] * y[b,c]      (== mean of gated x, gate const over HW)
// ---------------------------------------------------------------------------
__global__ __launch_bounds__(256) void k_eca(const float* __restrict__ y,
                                             const float* __restrict__ eca_w,
                                             float* __restrict__ gate,
                                             float* __restrict__ p) {
  const int i = blockIdx.x * blockDim.x + threadIdx.x;  // b*NC + c
  if (i >= NB * NC) return;
  const int c = i & (NC - 1);
  const float yl = (c > 0) ? y[i - 1] : 0.f;
  const float y0 = y[i];
  const float yr = (c < NC - 1) ? y[i + 1] : 0.f;
  const float z = eca_w[0] * yl + eca_w[1] * y0 + eca_w[2] * yr;
  const float g = 1.f / (1.f + expf(-z));
  gate[i] = g;
  p[i] = g * y0;
}

// ---------------------------------------------------------------------------
// Kernel 3: attention MLP (WMMA fp32), softmax, aggregated depthwise weights
// with the ECA gate folded in:
//   hdd    = gelu(p @ W1 + b1)                (16 x 256)   -> v_wmma_f32_16x16x4_f32
//   logits = hdd @ W2 + b2                    (16 x 12)
//   alpha  = softmax(logits)
//   aggw[b,c,ij] = gate[b,c] * sum_k alpha[b,k] * kernels_w[k,c,ij]
// One block, 512 threads = 16 waves; wave w owns output N-tile [16w, 16w+16).
// ---------------------------------------------------------------------------
__global__ __launch_bounds__(512) void k_mlp_aggw(
    const float* __restrict__ p, const float* __restrict__ w1,
    const float* __restrict__ b1, const float* __restrict__ w2,
    const float* __restrict__ b2, const float* __restrict__ kernels_w,
    const float* __restrict__ gate, float* __restrict__ aggw) {
  __shared__ float sP[16 * 256];
  __shared__ float sH[16 * 256];
  __shared__ float sAl[16 * 16];  // logits, then alphas (padded to 16)

  const int t = threadIdx.x;
  for (int i = t; i < 16 * 256; i += 512) sP[i] = p[i];
  __syncthreads();

  // ---- GEMM1 via V_WMMA_F32_16X16X4_F32, M=16 (batch), K=256, N tile = 16
  const int wv = t >> 5;          // wave id 0..15
  const int lane = t & 31;
  const int n0 = wv << 4;
  const int m = lane & 15;        // A-matrix row for this lane
  const int kg = (lane >> 4) << 1;  // K sub-group: 0 (lanes 0-15), 2 (lanes 16-31)
  const int n = n0 + (lane & 15); // B/C/D column for this lane

  v8f acc = {};
  #pragma unroll 8
  for (int k0 = 0; k0 < 256; k0 += 4) {
    v2f a, b;
    a.x = sP[m * 256 + k0 + kg];
    a.y = sP[m * 256 + k0 + kg + 1];
    b.x = w1[(size_t)(k0 + kg) * NC + n];
    b.y = w1[(size_t)(k0 + kg + 1) * NC + n];
    acc = __builtin_amdgcn_wmma_f32_16x16x4_f32(false, a, false, b,
                                                (short)0, acc, false, false);
  }

  // Epilogue: bias + exact GELU, scatter D tile (VGPR r holds rows r and r+8)
  const float bias = b1[n];
  #pragma unroll
  for (int r = 0; r < 8; ++r) {
    const int mr = r + ((lane >> 4) << 3);
    const float v = acc[r] + bias;
    sH[mr * 256 + n] = 0.5f * v * (1.f + erff(v * 0.70710678118654752f));
  }
  __syncthreads();

  // ---- GEMM2 (16x256 @ 256x12): scalar, one thread per (b, k)
  if (t < NB * NK) {
    const int bb = t / NK, kk = t - bb * NK;
    float s = b2[kk];
    for (int j = 0; j < NC; ++j) s += sH[bb * 256 + j] * w2[j * NK + kk];
    sAl[bb * 16 + kk] = s;
  }
  __syncthreads();

  // ---- softmax over K=12 per batch row
  if (t < NB) {
    float mx = -3.0e38f;
    for (int kk = 0; kk < NK; ++kk) mx = fmaxf(mx, sAl[t * 16 + kk]);
    float e[NK], s = 0.f;
    for (int kk = 0; kk < NK; ++kk) { e[kk] = expf(sAl[t * 16 + kk] - mx); s += e[kk]; }
    const float inv = 1.f / s;
    for (int kk = 0; kk < NK; ++kk) sAl[t * 16 + kk] = e[kk] * inv;
  }
  __syncthreads();

  // ---- aggregated depthwise weights with gate folded in
  for (int idx = t; idx < NB * NC * 9; idx += 512) {
    const int bb = idx / (NC * 9);
    const int rem = idx - bb * NC * 9;
    const int c = rem / 9, ij = rem - c * 9;
    float s = 0.f;
    #pragma unroll
    for (int kk = 0; kk < NK; ++kk)
      s += sAl[bb * 16 + kk] * kernels_w[(size_t)kk * NC * 9 + c * 9 + ij];
    aggw[idx] = s * gate[bb * NC + c];
  }
}

// ---------------------------------------------------------------------------
// Kernel 4: fused depthwise 3x3 conv (on raw x, gate folded into weights)
//           + channel LayerNorm + GELU + residual.
// Block = 256 threads = 256 pixels (2 rows x 128); thread owns one pixel for
// ALL channels -> mean/var are thread-local, conv results spill to 256 KB LDS
// (CDNA5 320 KB WGP LDS). No barriers in the hot path. Output stores are
// nontemporal (write-once stream) so they don't evict x halo rows from L2.
// ---------------------------------------------------------------------------
__global__ __launch_bounds__(256) void k_conv_ln(
    const float* __restrict__ x, const float* __restrict__ aggw,
    const float* __restrict__ gamma, const float* __restrict__ beta,
    float* __restrict__ out) {
  extern __shared__ float sConv[];  // [NC][256], 256 KB
  const int t = threadIdx.x;
  const int blk = blockIdx.x;
  const int b = blk >> 6;               // 64 blocks (2-row tiles) per batch
  const int h0 = (blk & 63) << 1;
  const int hh = t >> 7;                // 0..1
  const int w = t & 127;
  const int h = h0 + hh;

  const float* xb = x + (size_t)b * NC * HWPIX;
  const float* wb = aggw + (size_t)b * NC * 9;

  float sum = 0.f, sumsq = 0.f;
  for (int c = 0; c < NC; ++c) {
    const float* xc = xb + (size_t)c * HWPIX;
    const float* wc = wb + c * 9;
    if (c + 1 < NC)  // pull next channel's rows toward the caches
      __builtin_prefetch(xc + HWPIX + (size_t)h * NW, 0, 0);
    float acc = 0.f;
    #pragma unroll
    for (int i = 0; i < 3; ++i) {
      const int hr = h + i - 1;
      if (hr >= 0 && hr < NH) {
        const float* row = xc + (size_t)hr * NW;
        const float xm = (w > 0) ? row[w - 1] : 0.f;
        const float x0 = row[w];
        const float xp = (w < NW - 1) ? row[w + 1] : 0.f;
        acc += wc[i * 3 + 0] * xm + wc[i * 3 + 1] * x0 + wc[i * 3 + 2] * xp;
      }
    }
    sConv[c * 256 + t] = acc;
    sum += acc;
    sumsq += acc * acc;
  }

  const float mu = sum * (1.f / NC);
  const float var = sumsq * (1.f / NC) - mu * mu;
  const float rstd = rsqrtf(var + 1e-6f);

  const size_t pixoff = (size_t)h * NW + w;
  for (int c = 0; c < NC; ++c) {
    const float v = (sConv[c * 256 + t] - mu) * rstd * gamma[c] + beta[c];
    const float g = 0.5f * v * (1.f + erff(v * 0.70710678118654752f));
    // residual read stays cached (same rows as conv phase); store is NT
    const float r = g + xb[(size_t)c * HWPIX + pixoff];
    __builtin_nontemporal_store(r, &out[((size_t)b * NC + c) * HWPIX + pixoff]);
  }
}

// ---------------------------------------------------------------------------
extern "C" void kernel_launch(void* const* d_in, const int* in_sizes, int n_in,
                              void* d_out, int out_size, void* d_ws, size_t ws_size,
                              hipStream_t stream) {
  const float* x         = (const float*)d_in[0];
  const float* att_w1    = (const float*)d_in[1];
  const float* att_b1    = (const float*)d_in[2];
  const float* att_w2    = (const float*)d_in[3];
  const float* att_b2    = (const float*)d_in[4];
  const float* kernels_w = (const float*)d_in[5];
  const float* eca_w     = (const float*)d_in[6];
  const float* ln_gamma  = (const float*)d_in[7];
  const float* ln_beta   = (const float*)d_in[8];
  float* out = (float*)d_out;

  float* ws   = (float*)d_ws;
  float* y    = ws;                 // NB*NC
  float* gate = ws + NB * NC;       // NB*NC
  float* p    = ws + 2 * NB * NC;   // NB*NC
  float* aggw = ws + 3 * NB * NC;   // NB*NC*9

  k_gap<<<NB * NC, 256, 0, stream>>>(x, y);
  k_eca<<<(NB * NC + 255) / 256, 256, 0, stream>>>(y, eca_w, gate, p);
  k_mlp_aggw<<<1, 512, 0, stream>>>(p, att_w1, att_b1, att_w2, att_b2,
                                    kernels_w, gate, aggw);
  k_conv_ln<<<NB * (NH / 2), 256, NC * 256 * sizeof(float), stream>>>(
      x, aggw, ln_gamma, ln_beta, out);
}